// DotGAT_42992622633739
// MI455X (gfx1250) — compile-verified
//
#include <hip/hip_runtime.h>
#include <math.h>

// ---------------------------------------------------------------------------
// Types for CDNA5 WMMA f32 16x16x4: A is 16x4 f32 (2 VGPR/lane), B is 4x16,
// C/D is 16x16 f32 (8 VGPR/lane).
// ---------------------------------------------------------------------------
typedef __attribute__((ext_vector_type(2))) float v2f;
typedef __attribute__((ext_vector_type(8))) float v8f;

// Monotonic-bits float atomic max (works for mixed signs, init with -inf).
__device__ __forceinline__ void atomicMaxFloat(float* addr, float value) {
  if (value >= 0.0f) {
    atomicMax((int*)addr, __float_as_int(value));
  } else {
    atomicMin((unsigned int*)addr, __float_as_uint(value));
  }
}

// ---------------------------------------------------------------------------
// C[M,N] = A[M,K] * B[K,N], all fp32 row-major, via V_WMMA_F32_16X16X4_F32.
// One wave computes one 16x16 tile of C. M,N multiples of 16; K multiple of 4.
// ISA layout (CDNA5 ISA §7.12.2):
//   A 16x4 :  lanes 0-15 -> K=0,1 in v0,v1 ; lanes 16-31 -> K=2,3
//   B 4x16 :  v0 = row K (lanes 0-15) / K+2 (lanes 16-31), v1 = K+1 / K+3
//   C 16x16:  VGPR j: lanes 0-15 -> M=j, N=lane ; lanes 16-31 -> M=j+8
// ---------------------------------------------------------------------------
__global__ __launch_bounds__(256) void gemm_f32_wmma(
    const float* __restrict__ A, const float* __restrict__ B,
    float* __restrict__ C, int M, int N, int K) {
  const int lane = threadIdx.x & 31;
  const int wave = threadIdx.x >> 5;
  const int wavesPerBlock = blockDim.x >> 5;
  const int tile = blockIdx.x * wavesPerBlock + wave;
  const int tilesN = N >> 4;
  const int numTiles = (M >> 4) * tilesN;
  if (tile >= numTiles) return;          // wave-uniform: EXEC stays all-1s

  const int tm = (tile / tilesN) << 4;
  const int tn = (tile % tilesN) << 4;
  const int r  = lane & 15;              // A row / B,C column within tile
  const int kk = (lane >> 4) << 1;       // 0 for lanes 0-15, 2 for lanes 16-31

  const float* Arow = A + (size_t)(tm + r) * K;

  v8f acc = {};
  for (int k0 = 0; k0 < K; k0 += 4) {
    v2f a, b;
    float2 av = *(const float2*)(Arow + k0 + kk);   // b64 load
    a.x = av.x;
    a.y = av.y;
    b.x = B[(size_t)(k0 + kk) * N + tn + r];
    b.y = B[(size_t)(k0 + kk + 1) * N + tn + r];
    acc = __builtin_amdgcn_wmma_f32_16x16x4_f32(
        /*neg_a=*/false, a, /*neg_b=*/false, b,
        /*c_mod=*/(short)0, acc, /*reuse_a=*/false, /*reuse_b=*/false);
  }

  const int rowBase = tm + ((lane >> 4) << 3);   // +8 for upper half-wave
#pragma unroll
  for (int j = 0; j < 8; ++j) {
    C[(size_t)(rowBase + j) * N + tn + r] = acc[j];
  }
}

// ---------------------------------------------------------------------------
// Edge softmax passes, wave-per-edge. feat is [Nnodes, H*D] row-major with
// H*D == HD. Each lane owns PER_LANE = HD/32 contiguous floats of the row, so
// gathers are coalesced b128/b64 loads and scatter atomics are coalesced.
// ---------------------------------------------------------------------------
__global__ void fill_f32(float* __restrict__ p, float v, int n) {
  int i = blockIdx.x * blockDim.x + threadIdx.x;
  if (i < n) p[i] = v;
}

// Pass A: e[edge,h] = scale * dot(feat[src,h,:], feat[dst,h,:]);
//         atomic-max into m[dst,h]. One wave per edge.
template <int HD, int D>
__global__ __launch_bounds__(256) void edge_scores_wave(
    const float* __restrict__ feat, const int* __restrict__ src,
    const int* __restrict__ dst, float* __restrict__ e, float* __restrict__ m,
    int E, float scale) {
  constexpr int PER_LANE = HD / 32;          // 4 (layer1) or 2 (layer2)
  constexpr int LPH = D / PER_LANE;          // lanes per head: 4 or 32
  constexpr int H = HD / D;
  const int lane = threadIdx.x & 31;
  const int edge = (blockIdx.x * blockDim.x + threadIdx.x) >> 5;
  if (edge >= E) return;                     // wave-uniform
  const int s = src[edge], d = dst[edge];
  const float* fs = feat + (size_t)s * HD + lane * PER_LANE;
  const float* fd = feat + (size_t)d * HD + lane * PER_LANE;

  float acc;
  if constexpr (PER_LANE == 4) {
    float4 a = *(const float4*)fs;           // global_load_b128, coalesced
    float4 b = *(const float4*)fd;
    acc = a.x * b.x + a.y * b.y + a.z * b.z + a.w * b.w;
  } else {
    float2 a = *(const float2*)fs;           // global_load_b64, coalesced
    float2 b = *(const float2*)fd;
    acc = a.x * b.x + a.y * b.y;
  }
  // reduce within each head's lane group
#pragma unroll
  for (int off = 1; off < LPH; off <<= 1) acc += __shfl_xor(acc, off, 32);
  acc *= scale;

  if ((lane % LPH) == 0) {
    const int h = lane / LPH;
    e[(size_t)edge * H + h] = acc;
    atomicMaxFloat(m + (size_t)d * H + h, acc);
  }
}

// Pass B: e[idx] = exp(e - m[dst]); atomic-add into denom[dst,h].
// Thread per (edge,head); e is contiguous so access is already coalesced.
__global__ void edge_exp(float* __restrict__ e, const int* __restrict__ dst,
                         const float* __restrict__ m, float* __restrict__ denom,
                         int E, int H) {
  int idx = blockIdx.x * blockDim.x + threadIdx.x;
  if (idx >= E * H) return;
  int edge = idx / H;
  int h    = idx - edge * H;
  int d = dst[edge];
  float ex = __expf(e[idx] - m[(size_t)d * H + h]);
  e[idx] = ex;
  atomicAdd(denom + (size_t)d * H + h, ex);
}

// Pass C: out[dst,h,:] += (ex / max(denom,1e-9)) * feat[src,h,:]
// One wave per edge: coalesced gather + coalesced scatter atomics.
template <int HD, int D>
__global__ __launch_bounds__(256) void edge_agg_wave(
    const float* __restrict__ feat, const int* __restrict__ src,
    const int* __restrict__ dst, const float* __restrict__ ex,
    const float* __restrict__ denom, float* __restrict__ out, int E) {
  constexpr int PER_LANE = HD / 32;
  constexpr int LPH = D / PER_LANE;
  constexpr int H = HD / D;
  const int lane = threadIdx.x & 31;
  const int edge = (blockIdx.x * blockDim.x + threadIdx.x) >> 5;
  if (edge >= E) return;
  const int s = src[edge], d = dst[edge];
  const int h = lane / LPH;
  const float a =
      ex[(size_t)edge * H + h] / fmaxf(denom[(size_t)d * H + h], 1e-9f);
  const float* fs = feat + (size_t)s * HD + lane * PER_LANE;
  float* o = out + (size_t)d * HD + lane * PER_LANE;
  if constexpr (PER_LANE == 4) {
    float4 f = *(const float4*)fs;
    atomicAdd(o + 0, a * f.x);
    atomicAdd(o + 1, a * f.y);
    atomicAdd(o + 2, a * f.z);
    atomicAdd(o + 3, a * f.w);
  } else {
    float2 f = *(const float2*)fs;
    atomicAdd(o + 0, a * f.x);
    atomicAdd(o + 1, a * f.y);
  }
}

__global__ void elu_inplace(float* __restrict__ x, int n) {
  int i = blockIdx.x * blockDim.x + threadIdx.x;
  if (i < n) {
    float v = x[i];
    x[i] = v > 0.0f ? v : expm1f(v);
  }
}

// ---------------------------------------------------------------------------
// Launcher
// ---------------------------------------------------------------------------
static inline int cdiv_i(long long a, long long b) { return (int)((a + b - 1) / b); }

extern "C" void kernel_launch(void* const* d_in, const int* in_sizes, int n_in,
                              void* d_out, int out_size, void* d_ws, size_t ws_size,
                              hipStream_t stream) {
  const float* hIn = (const float*)d_in[0];   // [N, 128]
  const float* W1  = (const float*)d_in[1];   // [128, 128]
  const float* W2  = (const float*)d_in[2];   // [128, 64]
  const int*   src = (const int*)d_in[3];     // [E]
  const int*   dst = (const int*)d_in[4];     // [E]

  const int IN_DIM = 128, HEADS = 8, OUT_DIM = 64;
  const int HD1 = 128;                        // HEADS * HIDDEN
  const int Nn  = in_sizes[0] / IN_DIM;       // 50000 (multiple of 16)
  const int E   = in_sizes[3];                // 1600000

  // Workspace layout (floats). feat2 reuses feat1; layer-2 scores reuse scores.
  float* feat1  = (float*)d_ws;                     // Nn*128
  float* scores = feat1 + (size_t)Nn * HD1;         // E*HEADS
  float* mbuf   = scores + (size_t)E * HEADS;       // Nn*HEADS
  float* denom  = mbuf + (size_t)Nn * HEADS;        // Nn*HEADS
  float* h1     = denom + (size_t)Nn * HEADS;       // Nn*128

  const int TPB = 256;
  const int wavesPerBlock = TPB / 32;

  // ---------------- Layer 1 (H=8, D=16, scale=16^-0.5=0.25) ----------------
  {
    int tiles = (Nn / 16) * (HD1 / 16);
    gemm_f32_wmma<<<cdiv_i(tiles, wavesPerBlock), TPB, 0, stream>>>(
        hIn, W1, feat1, Nn, HD1, IN_DIM);
  }
  fill_f32<<<cdiv_i((long long)Nn * HEADS, TPB), TPB, 0, stream>>>(
      mbuf, -INFINITY, Nn * HEADS);
  hipMemsetAsync(denom, 0, (size_t)Nn * HEADS * sizeof(float), stream);
  hipMemsetAsync(h1, 0, (size_t)Nn * HD1 * sizeof(float), stream);

  edge_scores_wave<128, 16><<<cdiv_i(E, wavesPerBlock), TPB, 0, stream>>>(
      feat1, src, dst, scores, mbuf, E, 0.25f);
  edge_exp<<<cdiv_i((long long)E * HEADS, TPB), TPB, 0, stream>>>(
      scores, dst, mbuf, denom, E, HEADS);
  edge_agg_wave<128, 16><<<cdiv_i(E, wavesPerBlock), TPB, 0, stream>>>(
      feat1, src, dst, scores, denom, h1, E);

  elu_inplace<<<cdiv_i((long long)Nn * HD1, TPB), TPB, 0, stream>>>(h1, Nn * HD1);

  // ---------------- Layer 2 (H=1, D=64, scale=64^-0.5=0.125) ---------------
  float* feat2 = feat1;   // Nn*64 fits in the Nn*128 region
  {
    int tiles = (Nn / 16) * (OUT_DIM / 16);
    gemm_f32_wmma<<<cdiv_i(tiles, wavesPerBlock), TPB, 0, stream>>>(
        h1, W2, feat2, Nn, OUT_DIM, HD1);
  }
  fill_f32<<<cdiv_i(Nn, TPB), TPB, 0, stream>>>(mbuf, -INFINITY, Nn);
  hipMemsetAsync(denom, 0, (size_t)Nn * sizeof(float), stream);
  hipMemsetAsync(d_out, 0, (size_t)Nn * OUT_DIM * sizeof(float), stream);

  edge_scores_wave<64, 64><<<cdiv_i(E, wavesPerBlock), TPB, 0, stream>>>(
      feat2, src, dst, scores, mbuf, E, 0.125f);
  edge_exp<<<cdiv_i(E, TPB), TPB, 0, stream>>>(scores, dst, mbuf, denom, E, 1);
  edge_agg_wave<64, 64><<<cdiv_i(E, wavesPerBlock), TPB, 0, stream>>>(
      feat2, src, dst, scores, denom, (float*)d_out, E);
  // out2.mean(axis=1) with 1 head == identity -> d_out already holds [N, 64].
}